// AdaptiveGraphConvolution_47510928229054
// MI455X (gfx1250) — compile-verified
//
#include <hip/hip_runtime.h>

// Problem constants (match reference)
#define NN 50000      // nodes
#define NE 800000     // edges per graph
#define NG 4          // graphs
#define DD 128        // feature dim (in == out)

typedef __attribute__((ext_vector_type(2))) float v2f;
typedef __attribute__((ext_vector_type(8))) float v8f;

// ---------------------------------------------------------------------------
// out[n, f] = bias[f]   (vectorized float4; 50000*128/4 = 1.6M float4)
// ---------------------------------------------------------------------------
__global__ __launch_bounds__(256) void init_out_kernel(float* __restrict__ out,
                                                       const float* __restrict__ bias) {
    const float4* b4 = (const float4*)bias;
    float4* o4 = (float4*)out;
    unsigned idx = blockIdx.x * 256u + threadIdx.x;   // < NN*DD/4
    o4[idx] = b4[idx & 31u];                          // DD/4 == 32
}

// ---------------------------------------------------------------------------
// tmp = 0
// ---------------------------------------------------------------------------
__global__ __launch_bounds__(256) void zero_tmp_kernel(float* __restrict__ tmp) {
    float4 z = make_float4(0.f, 0.f, 0.f, 0.f);
    unsigned idx = blockIdx.x * 256u + threadIdx.x;
    ((float4*)tmp)[idx] = z;
}

// ---------------------------------------------------------------------------
// SpMM with mix folded in:  tmp[row] += (mix_l * val) * x[col]
// One wave per edge; each of the 32 lanes handles a float4 (128 feats).
// x is 25.6 MB -> resident in MI455X's 192 MB L2, so the random gather is an
// L2-bandwidth problem, not an HBM one. Exactly 8000 waves -> a static
// 100-iteration loop the compiler can unroll to keep >=2 gathers in flight.
// ---------------------------------------------------------------------------
#define SPMM_BLOCKS 1000
#define SPMM_WAVES  (SPMM_BLOCKS * 8)     // 8000; NE / 8000 == 100 exactly

__global__ __launch_bounds__(256) void spmm_kernel(const float* __restrict__ x,
                                                   const int* __restrict__ rows,
                                                   const int* __restrict__ cols,
                                                   const float* __restrict__ vals,
                                                   const float* __restrict__ mix,
                                                   int graph,
                                                   float* __restrict__ tmp) {
    const int lane = threadIdx.x & 31;
    const int wave = blockIdx.x * 8 + (threadIdx.x >> 5);
    const float m = mix[graph];
    const float4* x4 = (const float4*)x;

    #pragma unroll 2
    for (int i = 0; i < NE / SPMM_WAVES; ++i) {
        const int e   = wave + i * SPMM_WAVES;
        const int r   = rows[e];            // wave-uniform broadcast load
        const int c   = cols[e];
        const float s = vals[e] * m;
        float4 g = x4[c * 32 + lane];       // coalesced 512B per wave, L2 hit
        float* dst = tmp + (size_t)r * DD + lane * 4;
        atomicAdd(dst + 0, g.x * s);
        atomicAdd(dst + 1, g.y * s);
        atomicAdd(dst + 2, g.z * s);
        atomicAdd(dst + 3, g.w * s);
    }
}

// ---------------------------------------------------------------------------
// out += tmp @ Wg   via V_WMMA_F32_16X16X4_F32 (full fp32 matrix pipe).
//
// W is staged in LDS *K-pair packed*: ldsW[p][n] = {W[2p][n], W[2p+1][n]}
// (float2). A B-fragment for (kb, tile t) is then ONE aligned ds_load_b64
// into an even VGPR pair -> no v_dual_mov repacking before the WMMAs, and
// adjacent tiles merge into ds_load_2addr_b64.
//
// Bank analysis (64 banks x 4B): pair-row stride = PITCH2 float2 = 2*PITCH2
// floats; 2*144 = 288 ≡ 32 (mod 64), so lanes 0-15 (pair row p) and lanes
// 16-31 (pair row p+1) cover disjoint 32-bank windows -> conflict-free.
//
// WMMA f32 16x16x4 VGPR layouts (ISA 7.12.2):
//   A (16x4): lane = (half<<4)|m ; v0=A[m][2*half], v1=A[m][2*half+1]
//   B (4x16): v0 lanes0-15 = K row k, lanes16-31 = K row k+2; v1 = k+1 / k+3
//   C/D     : vgpr r: lanes0-15 -> M=r, N=lane; lanes16-31 -> M=r+8, N=lane-16
// ---------------------------------------------------------------------------
#define PITCH2 144   // float2 per k-pair row (128 data + 16 pad)

__global__ __launch_bounds__(256) void gemm_acc_kernel(const float* __restrict__ tmp,
                                                       const float* __restrict__ Wg,
                                                       float* __restrict__ out) {
    __shared__ v2f ldsW[(DD / 2) * PITCH2];   // 64*144*8B = 72 KB of 320 KB

    // Stage W (row-major [k][n]) into K-pair-packed LDS. One-time 64 KB.
    for (int i = threadIdx.x; i < (DD / 2) * DD; i += 256) {
        const int p = i >> 7;                 // k-pair index 0..63
        const int n = i & 127;                // column
        v2f w;
        w[0] = Wg[(2 * p)     * DD + n];
        w[1] = Wg[(2 * p + 1) * DD + n];
        ldsW[p * PITCH2 + n] = w;
    }
    __syncthreads();

    const int lane  = threadIdx.x & 31;
    const int waveI = threadIdx.x >> 5;
    const int m0 = (blockIdx.x * 8 + waveI) * 16;
    if (m0 >= NN) return;                     // wave-uniform: EXEC stays all-1s

    const int mrow  = lane & 15;              // M (A) / N (B,C) sub-index
    const int half  = lane >> 4;              // 0 or 1
    const int khalf = half * 2;               // A/B K sub-offset
    const int hi8   = half * 8;               // C/D row offset for upper lanes

    // C preload from out (read-modify-write, exclusive tile ownership).
    v8f acc[8];
    #pragma unroll
    for (int t = 0; t < 8; ++t)
        #pragma unroll
        for (int r = 0; r < 8; ++r)
            acc[t][r] = out[(size_t)(m0 + r + hi8) * DD + t * 16 + mrow];

    const float* arow = tmp + (size_t)(m0 + mrow) * DD;

    for (int k = 0; k < DD; k += 4) {
        const int kb = k + khalf;                             // even -> 8B aligned
        v2f a = *(const v2f*)(arow + kb);                     // A frag: global_load_b64
        const v2f* brow = &ldsW[(k / 2 + half) * PITCH2 + mrow];
        #pragma unroll
        for (int t = 0; t < 8; ++t) {
            v2f b = brow[t * 16];                             // ds_load_b64, aligned pair
            acc[t] = __builtin_amdgcn_wmma_f32_16x16x4_f32(
                /*neg_a=*/false, a, /*neg_b=*/false, b,
                /*c_mod=*/(short)0, acc[t],
                /*reuse_a=*/false, /*reuse_b=*/false);
        }
    }

    #pragma unroll
    for (int t = 0; t < 8; ++t)
        #pragma unroll
        for (int r = 0; r < 8; ++r)
            out[(size_t)(m0 + r + hi8) * DD + t * 16 + mrow] = acc[t][r];
}

// ---------------------------------------------------------------------------
// Launch: out=bias; for each graph: tmp=0 -> spmm -> wmma accumulate.
// All on `stream`; the in-order stream serializes the tmp reuse across graphs.
// ---------------------------------------------------------------------------
extern "C" void kernel_launch(void* const* d_in, const int* in_sizes, int n_in,
                              void* d_out, int out_size, void* d_ws, size_t ws_size,
                              hipStream_t stream) {
    (void)in_sizes; (void)n_in; (void)out_size; (void)ws_size;
    const float* x     = (const float*)d_in[0];
    const int*   erows = (const int*)d_in[1];
    const int*   ecols = (const int*)d_in[2];
    const float* evals = (const float*)d_in[3];
    const float* W     = (const float*)d_in[4];
    const float* mix   = (const float*)d_in[5];
    const float* bias  = (const float*)d_in[6];
    float* out = (float*)d_out;
    float* tmp = (float*)d_ws;              // 50000*128*4 = 25.6 MB scratch

    const int vec_blocks  = (NN * DD / 4) / 256;         // 6250, exact
    const int gemm_blocks = (NN / 16 + 7) / 8;           // 391 (3125 tiles)

    init_out_kernel<<<vec_blocks, 256, 0, stream>>>(out, bias);

    for (int l = 0; l < NG; ++l) {
        zero_tmp_kernel<<<vec_blocks, 256, 0, stream>>>(tmp);
        spmm_kernel<<<SPMM_BLOCKS, 256, 0, stream>>>(x,
                                                     erows + (size_t)l * NE,
                                                     ecols + (size_t)l * NE,
                                                     evals + (size_t)l * NE,
                                                     mix, l, tmp);
        gemm_acc_kernel<<<gemm_blocks, 256, 0, stream>>>(tmp, W + (size_t)l * DD * DD, out);
    }
}